// SurrogateModel_83124797047732
// MI455X (gfx1250) — compile-verified
//
#include <hip/hip_runtime.h>
#include <hip/hip_bf16.h>
#include <math.h>

// ---------------------------------------------------------------------------
// Surrogate gravitational-waveform model, CDNA5 (gfx1250, wave32).
//
// Pipeline (all on `stream`):
//   K1  coeff_kernel : C[22,128] = W@params + b      (phase row negated)
//   K1b harm_kernel  : swsh(l,m,theta) (f64, Goldberg) + exp(i*m*theta) tables
//   K2  gemv_wmma    : ys[22,8192] via V_WMMA_F32_16X16X4_F32 (mem-bound GEMV)
//   K2b t3_kernel    : ys[phase] += T3 post-Newtonian phase
//   K3a spline_fwd   : chunked-parallel Thomas forward elimination
//   K3b spline_bwd   : chunked-parallel back substitution -> Ms[22,8192]
//   K4  eval_kernel  : 1M spline evals + complex mode sum -> out (re,im)
//
// Spline parallelization: uniform knots => cp_i = 1/(4-cp_{i-1}) -> 2-sqrt(3)
// within ~30 fp32 steps; both scans are contractive with ratio ~0.268, so a
// 32-step warm-up from zero makes chunk-parallel scans exact to fp32 (error
// factor 0.268^32 ~ 5e-19).
// ---------------------------------------------------------------------------

typedef __attribute__((ext_vector_type(2))) float v2f;
typedef __attribute__((ext_vector_type(8))) float v8f;

#define NS 8192          // N_SAMPLE
#define NROWS 22         // 10 real + 10 imag + amp + phase
#define NNODES 128
#define NDIM 7
#define CHUNK 256
#define NCHUNK 32        // 32*256 = 8192 >= NS-2
#define WARM 32

// workspace layout (float offsets)
#define C_OFF    0               // [22*128]
#define HARM_OFF 2816            // [11]  (0 = Y22 for h22, 1..10 = NO22 modes)
#define ROTC_OFF 2832            // [10]  cos(m*theta)
#define ROTS_OFF 2848            // [10]  sin(m*theta)
#define YS_OFF   4096            // [22*8192]
#define MS_OFF   (YS_OFF + NROWS*NS)   // [22*8192]
#define DP_OFF   (MS_OFF + NROWS*NS)   // [22*8192] scratch for Thomas dp

// ----------------------------- K1: coefficients -----------------------------
__global__ void coeff_kernel(const float* __restrict__ params,
                             const float* __restrict__ Wr, const float* __restrict__ br,
                             const float* __restrict__ Wi, const float* __restrict__ bi,
                             const float* __restrict__ Wa, const float* __restrict__ ba,
                             const float* __restrict__ Wp, const float* __restrict__ bp,
                             float* __restrict__ ws) {
  int gid = blockIdx.x * blockDim.x + threadIdx.x;
  if (gid >= NROWS * NNODES) return;
  int row = gid >> 7, nn = gid & 127;
  const float* W; float b; float sgn = 1.0f;
  if (row < 10)      { W = Wr + (row * NNODES + nn) * NDIM; b = br[row * NNODES + nn]; }
  else if (row < 20) { int r = row - 10;
                       W = Wi + (r * NNODES + nn) * NDIM;   b = bi[r * NNODES + nn]; }
  else if (row == 20){ W = Wa + nn * NDIM; b = ba[nn]; }
  else               { W = Wp + nn * NDIM; b = bp[nn]; sgn = -1.0f; } // phase negated
  float acc = b;
#pragma unroll
  for (int i = 0; i < NDIM; ++i) acc += W[i] * params[i];
  ws[C_OFF + gid] = sgn * acc;
}

// --------------------- K1b: swsh + rotation tables (f64) --------------------
__device__ double fact_d(int n) { double r = 1.0; for (int i = 2; i <= n; ++i) r *= i; return r; }
__device__ double comb_d(int n, int k) {
  if (k < 0 || k > n) return 0.0;
  double r = 1.0;
  for (int i = 1; i <= k; ++i) r = r * (double)(n - k + i) / (double)i;
  return r;
}
__device__ float swsh_dev(int l, int m, float theta) { // s = -2, phi = 0
  const double PI = 3.14159265358979323846;
  double pref = ((m & 1) ? -1.0 : 1.0) *
      sqrt((2.0 * l + 1.0) / (4.0 * PI) * fact_d(l + m) * fact_d(l - m) /
           (fact_d(l - 2) * fact_d(l + 2)));
  double c = cos(0.5 * (double)theta), sn = sin(0.5 * (double)theta);
  int r0 = (m + 2 > 0) ? (m + 2) : 0;
  int r1 = (l + 2 < l + m) ? (l + 2) : (l + m);
  double tot = 0.0;
  for (int r = r0; r <= r1; ++r) {
    double coef = comb_d(l + 2, r) * comb_d(l - 2, r - 2 - m) *
                  (((l - r) & 1) ? -1.0 : 1.0);          // (-1)^(l-r+2)
    tot += coef * pow(c, (double)(2 * r - 2 - m)) * pow(sn, (double)(2 * l - 2 * r + 2 + m));
  }
  return (float)(pref * tot);
}

__global__ void harm_kernel(const float* __restrict__ theta_p,
                            const int* __restrict__ m_mode,
                            float* __restrict__ ws) {
  const int L[11] = {2, 2, 2, 3, 3, 3, 3, 4, 4, 4, 5};
  const int M[11] = {2, 1, 0, 0, 1, 2, 3, 2, 3, 4, 5};
  int tid = threadIdx.x;
  float th = theta_p[0];
  if (tid < 11) ws[HARM_OFF + tid] = swsh_dev(L[tid], M[tid], th);
  if (tid >= 16 && tid < 26) {
    int j = tid - 16;
    float ang = (float)m_mode[j] * th;
    ws[ROTC_OFF + j] = cosf(ang);
    ws[ROTS_OFF + j] = sinf(ang);
  }
}

// -------------------------- K2: WMMA GEMV over basis -------------------------
// out[row, s] = sum_n C[row,n] * basis[row,n,s]
// A[M,K] = C[row, k0+K]  (replicated over M)   B[K,N] = basis[row, k0+K, s0+N]
// => D[M,N] = out[s0+N] for every M.   Two accumulators break the D->C chain.
__global__ void __launch_bounds__(128)
gemv_wmma_kernel(const float* __restrict__ basis_real,
                 const float* __restrict__ basis_imag,
                 const float* __restrict__ basis_amp,
                 const float* __restrict__ basis_phase,
                 float* __restrict__ ws) {
  int wid  = (blockIdx.x * blockDim.x + threadIdx.x) >> 5;
  int lane = threadIdx.x & 31;
  int row  = wid >> 9;          // 22 rows
  int tile = wid & 511;         // 512 tiles of 16 samples
  int s0   = tile << 4;

  const float* basis;
  if (row < 10)       basis = basis_real  + (size_t)row        * NNODES * NS;
  else if (row < 20)  basis = basis_imag  + (size_t)(row - 10) * NNODES * NS;
  else if (row == 20) basis = basis_amp;
  else                basis = basis_phase;

  const float* crow = ws + C_OFF + row * NNODES;
  const int koff = (lane >= 16) ? 2 : 0;
  const int n    = lane & 15;

  v8f acc0 = {0.f, 0.f, 0.f, 0.f, 0.f, 0.f, 0.f, 0.f};
  v8f acc1 = {0.f, 0.f, 0.f, 0.f, 0.f, 0.f, 0.f, 0.f};
#pragma unroll 4
  for (int k = 0; k < NNODES; k += 8) {
    v2f a0, b0, a1, b1;
    a0.x = crow[k + koff];
    a0.y = crow[k + koff + 1];
    a1.x = crow[k + 4 + koff];
    a1.y = crow[k + 4 + koff + 1];
    const float* bp0 = basis + (size_t)(k + koff) * NS + s0 + n;
    const float* bp1 = bp0 + 4 * NS;
    b0.x = bp0[0];
    b0.y = bp0[NS];
    b1.x = bp1[0];
    b1.y = bp1[NS];
    if (k + 16 < NNODES) __builtin_prefetch(basis + (size_t)(k + 16) * NS + s0 + n, 0, 1);
    acc0 = __builtin_amdgcn_wmma_f32_16x16x4_f32(false, a0, false, b0,
                                                 (short)0, acc0, false, false);
    acc1 = __builtin_amdgcn_wmma_f32_16x16x4_f32(false, a1, false, b1,
                                                 (short)0, acc1, false, false);
  }
  if (lane < 16) ws[YS_OFF + row * NS + s0 + n] = acc0[0] + acc1[0]; // D[M=0,N=lane]
}

// --------------------------- K2b: T3 phase addend ----------------------------
__global__ void t3_kernel(const float* __restrict__ params,
                          const float* __restrict__ sur,
                          float* __restrict__ ws) {
  int s = blockIdx.x * blockDim.x + threadIdx.x;
  if (s >= NS) return;
  float q = params[0];
  float eta = q / ((1.0f + q) * (1.0f + q));
  float t = sur[s];
  // 2/(eta*theta^5), theta = X^(-1/8)  ->  (2/eta)*X^(5/8)
  float inv2e = 2.0f / eta;
  float vraw = inv2e * powf(eta * (1000.0f - t) * 0.2f, 0.625f);
  float vcal = inv2e * powf(eta * 2000.0f * 0.2f, 0.625f);
  ws[YS_OFF + 21 * NS + s] += vraw - vcal;
}

// ------------- K3a: Thomas forward elimination (chunk-parallel) --------------
// dp_i = cp_i * (g_i - dp_{i-1}),  g_i = 6*(y[i+2]-2y[i+1]+y[i])/h^2
__global__ void spline_fwd_kernel(const float* __restrict__ sur,
                                  float* __restrict__ ws) {
  __shared__ float cps64[64];
  if (threadIdx.x == 0) {
    float cp = 0.0f;
    for (int i = 0; i < 64; ++i) { cp = 1.0f / (4.0f - cp); cps64[i] = cp; }
  }
  __syncthreads();
  int g = blockIdx.x * blockDim.x + threadIdx.x;
  int row = g >> 5, chunk = g & (NCHUNK - 1);
  if (row >= NROWS) return;
  float cinf = cps64[63];

  float x0 = sur[0], xN = sur[NS - 1];
  float h = (xN - x0) / (float)(NS - 1);
  float g6 = 6.0f / (h * h);

  const float* y = ws + YS_OFF + row * NS;
  float* dp = ws + DP_OFF + row * NS;

  int start = chunk * CHUNK;
  int end = start + CHUNK; if (end > NS - 2) end = NS - 2;
  int wlo = start - WARM;  if (wlo < 0) wlo = 0;

  float y0 = y[wlo], y1 = y[wlo + 1];
  float d = 0.0f;
  for (int i = wlo; i < start; ++i) {          // warm-up (no stores)
    float y2 = y[i + 2];
    d = cinf * ((y2 - 2.0f * y1 + y0) * g6 - d);   // wlo>=224 here => cp==cinf
    y0 = y1; y1 = y2;
  }
  for (int i = start; i < end; ++i) {          // main
    float y2 = y[i + 2];
    float cp = (i < 64) ? cps64[i] : cinf;
    d = cp * ((y2 - 2.0f * y1 + y0) * g6 - d);
    dp[i] = d;
    y0 = y1; y1 = y2;
  }
}

// --------------- K3b: back substitution (chunk-parallel) ---------------------
// M_i = dp_i - cp_i * M_{i+1};  Ms = [0, M_0..M_{NS-3}, 0]
__global__ void spline_bwd_kernel(float* __restrict__ ws) {
  __shared__ float cps64[64];
  if (threadIdx.x == 0) {
    float cp = 0.0f;
    for (int i = 0; i < 64; ++i) { cp = 1.0f / (4.0f - cp); cps64[i] = cp; }
  }
  __syncthreads();
  int g = blockIdx.x * blockDim.x + threadIdx.x;
  int row = g >> 5, chunk = g & (NCHUNK - 1);
  if (row >= NROWS) return;
  float cinf = cps64[63];

  const float* dp = ws + DP_OFF + row * NS;
  float* Ms = ws + MS_OFF + row * NS;

  int start = chunk * CHUNK;
  int end = start + CHUNK; if (end > NS - 2) end = NS - 2;
  int whi = end + WARM;    if (whi > NS - 2) whi = NS - 2;

  float m = 0.0f;
  for (int i = whi - 1; i >= end; --i)          // warm-up (cp==cinf: i>=256>64)
    m = dp[i] - cinf * m;
  for (int i = end - 1; i >= start; --i) {      // main
    float cp = (i < 64) ? cps64[i] : cinf;
    m = dp[i] - cp * m;
    Ms[i + 1] = m;
  }
  if (chunk == 0)   Ms[0] = 0.0f;               // natural BCs
  if (end == NS - 2) Ms[NS - 1] = 0.0f;
}

// ------------------- K4: spline evaluation + mode combination ----------------
__global__ void __launch_bounds__(256)
eval_kernel(const float* __restrict__ time_p,
            const float* __restrict__ sur,
            const float* __restrict__ ws,
            float* __restrict__ out, int N) {
  __shared__ float sh[32];  // [0..10]=harm, [11..20]=rotc, [21..30]=rots
  int tid = threadIdx.x;
  if (tid < 11)       sh[tid] = ws[HARM_OFF + tid];
  else if (tid < 21)  sh[tid] = ws[ROTC_OFF + tid - 11];
  else if (tid < 31)  sh[tid] = ws[ROTS_OFF + tid - 21];
  __syncthreads();

  int i = blockIdx.x * blockDim.x + tid;
  if (i >= N) return;

  float t = time_p[i];
  float x0 = sur[0], xN = sur[NS - 1];
  float inv_dx = (float)(NS - 1) / (xN - x0);

  int idx = (int)floorf((t - x0) * inv_dx);
  idx = idx < 0 ? 0 : (idx > NS - 2 ? NS - 2 : idx);
#pragma unroll
  for (int it = 0; it < 2; ++it) {            // fix fp rounding vs searchsorted
    if (idx > 0 && t < sur[idx]) --idx;
    else if (idx < NS - 2 && t >= sur[idx + 1]) ++idx;
  }
  float x_lo = sur[idx], x_hi = sur[idx + 1];
  float hh = x_hi - x_lo;
  float invh = 1.0f / hh;
  float A = (x_hi - t) * invh;
  float B = (t - x_lo) * invh;
  float A3 = A * A * A - A;
  float B3 = B * B * B - B;
  float hh6 = hh * hh * (1.0f / 6.0f);

  const float* ys = ws + YS_OFF;
  const float* Ms = ws + MS_OFF;

  float re = 0.0f, im = 0.0f;
#pragma unroll
  for (int m = 0; m < 10; ++m) {
    int ro = m * NS, io = (10 + m) * NS;
    float r  = A * ys[ro + idx] + B * ys[ro + idx + 1] +
               (A3 * Ms[ro + idx] + B3 * Ms[ro + idx + 1]) * hh6;
    float ii = A * ys[io + idx] + B * ys[io + idx + 1] +
               (A3 * Ms[io + idx] + B3 * Ms[io + idx + 1]) * hh6;
    float hm = sh[1 + m];
    float rc = sh[11 + m], rs = sh[21 + m];
    re += hm * (r * rc - ii * rs);
    im += hm * (r * rs + ii * rc);
  }
  int ao = 20 * NS, po = 21 * NS;
  float amp = A * ys[ao + idx] + B * ys[ao + idx + 1] +
              (A3 * Ms[ao + idx] + B3 * Ms[ao + idx + 1]) * hh6;
  float ph  = A * ys[po + idx] + B * ys[po + idx + 1] +
              (A3 * Ms[po + idx] + B3 * Ms[po + idx + 1]) * hh6;
  float sp, cp;
  sincosf(ph, &sp, &cp);
  float y22 = sh[0];
  re += amp * cp * y22;
  im += amp * sp * y22;

  out[2 * i]     = re;
  out[2 * i + 1] = im;
}

// --------------------------------- launcher ----------------------------------
extern "C" void kernel_launch(void* const* d_in, const int* in_sizes, int n_in,
                              void* d_out, int out_size, void* d_ws, size_t ws_size,
                              hipStream_t stream) {
  const float* time_p      = (const float*)d_in[0];
  const float* params      = (const float*)d_in[1];
  const float* theta       = (const float*)d_in[2];
  const float* sur         = (const float*)d_in[3];
  const float* basis_real  = (const float*)d_in[4];
  const float* W_real      = (const float*)d_in[5];
  const float* b_real      = (const float*)d_in[6];
  const float* basis_imag  = (const float*)d_in[7];
  const float* W_imag      = (const float*)d_in[8];
  const float* b_imag      = (const float*)d_in[9];
  const float* basis_amp   = (const float*)d_in[10];
  const float* W_amp       = (const float*)d_in[11];
  const float* b_amp       = (const float*)d_in[12];
  const float* basis_phase = (const float*)d_in[13];
  const float* W_phase     = (const float*)d_in[14];
  const float* b_phase     = (const float*)d_in[15];
  const int*   m_mode      = (const int*)d_in[16];
  float* ws  = (float*)d_ws;
  float* out = (float*)d_out;
  int N = in_sizes[0];

  coeff_kernel<<<11, 256, 0, stream>>>(params, W_real, b_real, W_imag, b_imag,
                                       W_amp, b_amp, W_phase, b_phase, ws);
  harm_kernel<<<1, 32, 0, stream>>>(theta, m_mode, ws);
  gemv_wmma_kernel<<<2816, 128, 0, stream>>>(basis_real, basis_imag,
                                             basis_amp, basis_phase, ws);
  t3_kernel<<<32, 256, 0, stream>>>(params, sur, ws);
  spline_fwd_kernel<<<3, 256, 0, stream>>>(sur, ws);
  spline_bwd_kernel<<<3, 256, 0, stream>>>(ws);
  eval_kernel<<<(N + 255) / 256, 256, 0, stream>>>(time_p, sur, ws, out, N);
}